// RepresentationSummarizerMultiInstance_74792560492688
// MI455X (gfx1250) — compile-verified
//
#include <hip/hip_runtime.h>

// out[b,l] = dot(s_s[b,l,:], W[l,0,:]) + bias[l]
// B=256, L=700, D=1024, O=1.  Memory-bound (0.5 FLOP/byte): stream s_s once
// (NT loads), keep W (2.87MB) L2-resident, compute with v_wmma_f32_16x16x4_f32
// using the diagonal of a 16(pos) x 16(pos) tile per batch.

typedef __attribute__((ext_vector_type(2))) float v2f;
typedef __attribute__((ext_vector_type(8))) float v8f;

#define B_DIM   256
#define L_DIM   700
#define D_DIM   1024
#define LTILES  44            // ceil(700/16)
#define KSTEPS  (D_DIM / 4)   // 256 WMMA K-steps of 4
#define KU      8             // unroll factor

__global__ __launch_bounds__(256)
void summarizer_wmma_kernel(const float* __restrict__ s_s,
                            const float* __restrict__ W,
                            const float* __restrict__ bias,
                            float* __restrict__ out)
{
    const int lane = threadIdx.x & 31;
    const int wave = threadIdx.x >> 5;
    const int task = blockIdx.x * 8 + wave;      // 0 .. 11263 (exact)
    const int b    = task / LTILES;              // batch index
    const int l0   = (task % LTILES) * 16;       // position tile base

    // Per-lane operand addressing (f32 16x4 A / 4x16 B WMMA layouts):
    //   lane < 16 : holds K = {0,1} of row (lane),     in VGPR0/VGPR1
    //   lane >= 16: holds K = {2,3} of row (lane&15),  in VGPR0/VGPR1
    // => one 8-byte contiguous load per operand per K-step.
    const int row = lane & 15;
    const int hi  = lane >> 4;                   // 0 -> K0..1, 1 -> K2..3
    int l = l0 + row;
    if (l >= L_DIM) l = L_DIM - 1;               // clamp last (ragged) tile

    const float* aPtr = s_s + ((size_t)b * L_DIM + l) * (size_t)D_DIM + hi * 2;
    const float* bPtr = W   + (size_t)l * (size_t)D_DIM + hi * 2;

    v8f c = {0.f, 0.f, 0.f, 0.f, 0.f, 0.f, 0.f, 0.f};

    for (int k = 0; k < KSTEPS; k += KU) {
        v2f av[KU];
        v2f wv[KU];
#pragma unroll
        for (int u = 0; u < KU; ++u) {
            // s_s streamed exactly once -> non-temporal; W stays RT (L2-hot).
            av[u] = __builtin_nontemporal_load((const v2f*)(aPtr + (k + u) * 4));
            wv[u] = *(const v2f*)(bPtr + (k + u) * 4);
        }
#pragma unroll
        for (int u = 0; u < KU; ++u) {
            // (neg_a, A, neg_b, B, c_mod, C, reuse_a, reuse_b)
            c = __builtin_amdgcn_wmma_f32_16x16x4_f32(
                    false, av[u], false, wv[u], (short)0, c, false, false);
        }
    }

    // C[m][n] layout: VGPR r -> lanes 0-15: M=r, N=lane ; lanes 16-31: M=8+r, N=lane-16.
    // Diagonal M==N lives in: lanes 0..7  -> c[lane]      (m = lane)
    //                         lanes 24..31-> c[lane-24]   (m = lane-16)
    float diag = 0.f;
#pragma unroll
    for (int r = 0; r < 8; ++r) {
        if (lane == r)      diag = c[r];
        if (lane == 24 + r) diag = c[r];
    }

    int m = -1;
    if (lane < 8)        m = lane;
    else if (lane >= 24) m = lane - 16;

    if (m >= 0) {
        const int lo = l0 + m;
        if (lo < L_DIM) {
            out[(size_t)b * L_DIM + lo] = diag + bias[lo];
        }
    }
}

extern "C" void kernel_launch(void* const* d_in, const int* in_sizes, int n_in,
                              void* d_out, int out_size, void* d_ws, size_t ws_size,
                              hipStream_t stream) {
    (void)in_sizes; (void)n_in; (void)out_size; (void)d_ws; (void)ws_size;
    const float* s_s  = (const float*)d_in[0];   // [B, L, D] f32
    const float* W    = (const float*)d_in[1];   // [L, 1, D] f32
    const float* bias = (const float*)d_in[2];   // [L, 1]    f32
    float* out = (float*)d_out;                  // [B, L, 1] f32

    const int tasks  = B_DIM * LTILES;           // 11264 waves
    const int blocks = tasks / 8;                // 8 waves (256 thr) per block
    hipLaunchKernelGGL(summarizer_wmma_kernel, dim3(blocks), dim3(256), 0, stream,
                       s_s, W, bias, out);
}